// GINEMB_66898410602746
// MI455X (gfx1250) — compile-verified
//
#include <hip/hip_runtime.h>

// ---------------------------------------------------------------------------
// GIN (3-layer, eps=0, mean aggregation) for MI455X / gfx1250.
// Bound by aggregation bandwidth (~5.6 GB total @ 23.3 TB/s); GEMM uses
// fp32 WMMA (v_wmma_f32_16x16x4_f32) for exact fp32 semantics at zero cost.
// ---------------------------------------------------------------------------

typedef __attribute__((ext_vector_type(2))) float v2f;
typedef __attribute__((ext_vector_type(8))) float v8f;

#define FEATS 128
#define LDSS  132   // padded LDS row stride (dwords): conflict-free b64 reads

// ---- zero a buffer (float4 granularity) -----------------------------------
__global__ __launch_bounds__(256) void k_zero_f4(float4* __restrict__ p, long n4) {
    long i = (long)blockIdx.x * 256 + threadIdx.x;
    if (i < n4) p[i] = float4{0.f, 0.f, 0.f, 0.f};
}

// ---- in-degree count -------------------------------------------------------
__global__ __launch_bounds__(256) void k_degree(const int* __restrict__ dst,
                                                float* __restrict__ deg, int n_edges) {
    int i = blockIdx.x * 256 + threadIdx.x;
    if (i < n_edges) atomicAdd(&deg[dst[i]], 1.0f);
}

__global__ __launch_bounds__(256) void k_inv_deg(float* __restrict__ deg, int n) {
    int i = blockIdx.x * 256 + threadIdx.x;
    if (i < n) deg[i] = 1.0f / fmaxf(deg[i], 1.0f);
}

// ---- scatter: agg[dst] += h[src]  (1 wave per edge, float4 per lane) ------
__global__ __launch_bounds__(256) void k_scatter(const float* __restrict__ h,
                                                 const int* __restrict__ src,
                                                 const int* __restrict__ dst,
                                                 float* __restrict__ agg, int n_edges) {
    int e    = (blockIdx.x * 256 + threadIdx.x) >> 5;   // edge = wave index
    int lane = threadIdx.x & 31;
    if (e >= n_edges) return;
    int s = src[e];
    int d = dst[e];
    // 32 lanes x 16B = one 512B feature row, fully coalesced
    float4 v = ((const float4*)(h + (size_t)s * FEATS))[lane];
    float* ap = agg + (size_t)d * FEATS + lane * 4;
    atomicAdd(ap + 0, v.x);
    atomicAdd(ap + 1, v.y);
    atomicAdd(ap + 2, v.z);
    atomicAdd(ap + 3, v.w);
}

// ---- fused GIN layer: out = act((h + inv_deg*agg) @ W + b) ----------------
// Block: 256 threads (8 waves). Block handles 16 rows x 128 output cols.
// Wave w computes the 16x16 tile at column block 16*w via f32 WMMA, K=128.
// Bias is folded into the WMMA C-accumulator initialization.
template <bool RELU>
__global__ __launch_bounds__(256) void k_gin_layer(const float* __restrict__ h,
                                                   const float* __restrict__ agg,
                                                   const float* __restrict__ inv_deg,
                                                   const float* __restrict__ W,
                                                   const float* __restrict__ bias,
                                                   float* __restrict__ out,
                                                   int n_nodes) {
    __shared__ float A[16][LDSS];

    const int tid       = threadIdx.x;
    const int block_row = blockIdx.x * 16;

    // Stage 1: build A-tile = h + inv_deg * agg  (16 rows x 128 cols in LDS).
    {
        int r   = tid >> 4;            // 0..15 (row in tile)
        int c   = (tid & 15) * 8;      // 8 floats per thread
        int row = block_row + r;
        if (row >= n_nodes) row = n_nodes - 1;     // N%16==0, safety clamp
        float id = inv_deg[row];
        const float4* hp = (const float4*)(h   + (size_t)row * FEATS + c);
        const float4* gp = (const float4*)(agg + (size_t)row * FEATS + c);
        float4 h0 = hp[0], h1 = hp[1];
        float4 g0 = gp[0], g1 = gp[1];
        A[r][c + 0] = h0.x + id * g0.x;
        A[r][c + 1] = h0.y + id * g0.y;
        A[r][c + 2] = h0.z + id * g0.z;
        A[r][c + 3] = h0.w + id * g0.w;
        A[r][c + 4] = h1.x + id * g1.x;
        A[r][c + 5] = h1.y + id * g1.y;
        A[r][c + 6] = h1.z + id * g1.z;
        A[r][c + 7] = h1.w + id * g1.w;
    }
    __syncthreads();

    // Stage 2: per-wave 16x16 WMMA tile, K = 128 in steps of 4.
    const int lane  = tid & 31;
    const int wv    = tid >> 5;              // 0..7 -> output column block
    const int n0    = wv * 16;
    const int m     = lane & 15;             // A-matrix row / B-matrix col
    const int khalf = (lane >> 4) << 1;      // lanes 0-15: K+{0,1}; 16-31: K+{2,3}
    const int ncol  = n0 + m;

    // Seed accumulator with bias: every output row of this column gets bias[ncol].
    const float bv = bias[ncol];
    v8f acc = {bv, bv, bv, bv, bv, bv, bv, bv};

    #pragma unroll 4
    for (int k0 = 0; k0 < FEATS; k0 += 4) {
        v2f a, b;
        // ISA 32-bit A 16x4 layout: v0=K(khalf), v1=K(khalf+1), M = lane&15
        a.x = A[m][k0 + khalf];
        a.y = A[m][k0 + khalf + 1];
        // mirrored B 4x16 layout: N = lane&15, K split in lane halves
        const float* wp = W + (size_t)(k0 + khalf) * FEATS + ncol;
        b.x = wp[0];
        b.y = wp[FEATS];
        acc = __builtin_amdgcn_wmma_f32_16x16x4_f32(
            /*neg_a=*/false, a, /*neg_b=*/false, b,
            /*c_mod=*/(short)0, acc, /*reuse_a=*/false, /*reuse_b=*/false);
    }

    // Epilogue per C/D layout: VGPR r -> M = r (lanes 0-15) / r+8 (lanes 16-31)
    const int orow0 = block_row + ((lane >> 4) << 3);
    float* op = out + (size_t)orow0 * FEATS + ncol;

    if (block_row + 16 <= n_nodes) {           // uniform fast path (N%16==0)
        #pragma unroll
        for (int r = 0; r < 8; ++r) {
            float v = acc[r];
            if (RELU) v = fmaxf(v, 0.0f);
            op[(size_t)r * FEATS] = v;
        }
    } else {                                   // uniform guarded tail path
        #pragma unroll
        for (int r = 0; r < 8; ++r) {
            float v = acc[r];
            if (RELU) v = fmaxf(v, 0.0f);
            if (orow0 + r < n_nodes) op[(size_t)r * FEATS] = v;
        }
    }
}

// ---------------------------------------------------------------------------
extern "C" void kernel_launch(void* const* d_in, const int* in_sizes, int n_in,
                              void* d_out, int out_size, void* d_ws, size_t ws_size,
                              hipStream_t stream) {
    const float* x      = (const float*)d_in[0];
    const float* Wt[3]  = {(const float*)d_in[1], (const float*)d_in[3], (const float*)d_in[5]};
    const float* bt[3]  = {(const float*)d_in[2], (const float*)d_in[4], (const float*)d_in[6]};
    const int*   esrc   = (const int*)d_in[7];
    const int*   edst   = (const int*)d_in[8];
    float*       out    = (float*)d_out;

    const int N = in_sizes[0] / FEATS;   // 100000
    const int E = in_sizes[7];           // 1600000

    // workspace layout: inv_deg[N] | agg[N*128] | hbuf[N*128]  (~103 MB)
    float* inv_deg = (float*)d_ws;
    float* agg     = inv_deg + N;
    float* hbuf    = agg + (size_t)N * FEATS;

    // ---- degree / inverse degree (recomputed every call: deterministic) ----
    k_zero_f4<<<(N / 4 + 255) / 256, 256, 0, stream>>>((float4*)inv_deg, N / 4);
    k_degree <<<(E + 255) / 256,     256, 0, stream>>>(edst, inv_deg, E);
    k_inv_deg<<<(N + 255) / 256,     256, 0, stream>>>(inv_deg, N);

    const long n4    = (long)N * FEATS / 4;
    const int  zgrid = (int)((n4 + 255) / 256);
    const int  sgrid = (E * 32 + 255) / 256;   // one wave per edge
    const int  ggrid = (N + 15) / 16;

    const float* hin = x;
    for (int l = 0; l < 3; ++l) {
        float* hout = (l == 2) ? out : hbuf;   // in-place on hbuf is safe
        k_zero_f4 <<<zgrid, 256, 0, stream>>>((float4*)agg, n4);
        k_scatter <<<sgrid, 256, 0, stream>>>(hin, esrc, edst, agg, E);
        if (l < 2)
            k_gin_layer<true ><<<ggrid, 256, 0, stream>>>(hin, agg, inv_deg,
                                                          Wt[l], bt[l], hout, N);
        else
            k_gin_layer<false><<<ggrid, 256, 0, stream>>>(hin, agg, inv_deg,
                                                          Wt[l], bt[l], hout, N);
        hin = hbuf;
    }
}